// NoisyPositionEmbedder_24412594111346
// MI455X (gfx1250) — compile-verified
//
#include <hip/hip_runtime.h>
#include <hip/hip_bf16.h>

// ---------------------------------------------------------------------------
// NoisyPositionEmbedder for MI455X (gfx1250, wave32, WMMA).
//
// Roofline reasoning: the naive path reads all of zij_trunk (512 MB). The
// gather only needs 128 blocks * 8 * 32 = 32768 unique token-pair rows of
// zij (16.8 MB). plm in+out is 67 MB. Total ~95 MB => ~4 us at 23.3 TB/s;
// compute (~0.2 GFLOP fp32) is negligible, done with V_WMMA_F32_16X16X4_F32
// to keep fp32 precision while exercising the matrix pipe.
// ---------------------------------------------------------------------------

#define N_TOK   1024
#define N_ATOM  4096
#define C_S     384
#define C_Z     128
#define C_ATOM  128
#define C_PAIR  16
#define N_QUERY 32
#define N_KEY   128
#define N_BLOCKS 128
#define EPSV    1e-5f

typedef float v2f __attribute__((ext_vector_type(2)));
typedef float v8f __attribute__((ext_vector_type(8)));

__device__ __forceinline__ float wave_sum(float v) {
#pragma unroll
  for (int off = 16; off >= 1; off >>= 1) v += __shfl_xor(v, off, 32);
  return v;
}

// ---------------------------------------------------------------------------
// Kernel A: s = (LN(si_trunk) * ln_s_w) @ W_s + b_s, masked by token_mask.
// Grid: 32 blocks x 256 threads. Each block: 32 rows x 128 cols.
// WMMA f32 16x16x4: per wave, fixed N-tile (n0 = wave*16), two M-tiles.
// ---------------------------------------------------------------------------
#define A_ST 387  // 384 + 3 (odd stride -> conflict-free A-fragment reads)

__global__ __launch_bounds__(256) void s_proj_kernel(
    const float* __restrict__ si, const float* __restrict__ ln_s_w,
    const float* __restrict__ W_s, const float* __restrict__ b_s,
    const float* __restrict__ token_mask, float* __restrict__ s_out) {
  __shared__ float Alds[32 * A_ST];  // 49.5 KB
  const int tid  = threadIdx.x;
  const int wave = tid >> 5;
  const int lane = tid & 31;
  const int half = lane >> 4;
  const int l16  = lane & 15;
  const int base = blockIdx.x * 32;

  // LN weights for this lane's 12 channels (c = lane*4 + 128*j)
  float4 w0 = *(const float4*)(ln_s_w + lane * 4);
  float4 w1 = *(const float4*)(ln_s_w + lane * 4 + 128);
  float4 w2 = *(const float4*)(ln_s_w + lane * 4 + 256);

  // Phase 1: layernorm 32 rows (wave-per-row, 4 rows per wave)
#pragma unroll
  for (int it = 0; it < 4; ++it) {
    const int r = wave * 4 + it;
    const int t = base + r;
    const float* row = si + (size_t)t * C_S;
    float4 x0 = *(const float4*)(row + lane * 4);
    float4 x1 = *(const float4*)(row + lane * 4 + 128);
    float4 x2 = *(const float4*)(row + lane * 4 + 256);
    float s  = x0.x + x0.y + x0.z + x0.w + x1.x + x1.y + x1.z + x1.w +
               x2.x + x2.y + x2.z + x2.w;
    float ss = x0.x*x0.x + x0.y*x0.y + x0.z*x0.z + x0.w*x0.w +
               x1.x*x1.x + x1.y*x1.y + x1.z*x1.z + x1.w*x1.w +
               x2.x*x2.x + x2.y*x2.y + x2.z*x2.z + x2.w*x2.w;
    s  = wave_sum(s);
    ss = wave_sum(ss);
    const float mu   = s * (1.f / C_S);
    const float var  = ss * (1.f / C_S) - mu * mu;
    const float rstd = rsqrtf(var + EPSV);
    float* dst = Alds + r * A_ST + lane * 4;
    dst[0]   = (x0.x - mu) * rstd * w0.x;
    dst[1]   = (x0.y - mu) * rstd * w0.y;
    dst[2]   = (x0.z - mu) * rstd * w0.z;
    dst[3]   = (x0.w - mu) * rstd * w0.w;
    dst[128] = (x1.x - mu) * rstd * w1.x;
    dst[129] = (x1.y - mu) * rstd * w1.y;
    dst[130] = (x1.z - mu) * rstd * w1.z;
    dst[131] = (x1.w - mu) * rstd * w1.w;
    dst[256] = (x2.x - mu) * rstd * w2.x;
    dst[257] = (x2.y - mu) * rstd * w2.y;
    dst[258] = (x2.z - mu) * rstd * w2.z;
    dst[259] = (x2.w - mu) * rstd * w2.w;
  }
  __syncthreads();

  // Phase 2: [32 x 384] @ [384 x 16] per wave (two 16x16 M-tiles)
  const int n0 = wave * 16;
  v8f acc0 = {0.f, 0.f, 0.f, 0.f, 0.f, 0.f, 0.f, 0.f};
  v8f acc1 = {0.f, 0.f, 0.f, 0.f, 0.f, 0.f, 0.f, 0.f};
  const float bsv = b_s[n0 + l16];
  for (int k0 = 0; k0 < C_S; k0 += 4) {
    const int kk = k0 + 2 * half;
    v2f bb;
    bb.x = W_s[(size_t)kk * C_ATOM + n0 + l16];
    bb.y = W_s[(size_t)(kk + 1) * C_ATOM + n0 + l16];
    v2f a0, a1;
    a0.x = Alds[l16 * A_ST + kk];
    a0.y = Alds[l16 * A_ST + kk + 1];
    a1.x = Alds[(16 + l16) * A_ST + kk];
    a1.y = Alds[(16 + l16) * A_ST + kk + 1];
    acc0 = __builtin_amdgcn_wmma_f32_16x16x4_f32(false, a0, false, bb,
                                                 (short)0, acc0, false, false);
    acc1 = __builtin_amdgcn_wmma_f32_16x16x4_f32(false, a1, false, bb,
                                                 (short)0, acc1, false, false);
  }
#pragma unroll
  for (int v = 0; v < 8; ++v) {
    const int m  = v + half * 8;
    const int t0 = base + m;
    const int t1 = base + 16 + m;
    s_out[(size_t)t0 * C_ATOM + n0 + l16] = (acc0[v] + bsv) * token_mask[t0];
    s_out[(size_t)t1 * C_ATOM + n0 + l16] = (acc1[v] + bsv) * token_mask[t1];
  }
}

// ---------------------------------------------------------------------------
// Kernel B: tok[a] = searchsorted(cumsum(num_atoms_per_token), a, 'right')
// Single block of 1024 threads: Hillis-Steele scan in LDS + binary search.
// ---------------------------------------------------------------------------
__global__ __launch_bounds__(1024) void tok_kernel(
    const int* __restrict__ napt, int* __restrict__ tok) {
  __shared__ int buf[2][N_TOK];
  const int tid = threadIdx.x;
  buf[0][tid] = napt[tid];
  __syncthreads();
  int src = 0;
  for (int off = 1; off < N_TOK; off <<= 1) {
    int v = buf[src][tid];
    if (tid >= off) v += buf[src][tid - off];
    buf[1 - src][tid] = v;
    src ^= 1;
    __syncthreads();
  }
  for (int a = tid; a < N_ATOM; a += N_TOK) {
    int lo = 0, hi = N_TOK - 1, res = N_TOK - 1;
    while (lo <= hi) {
      const int mid = (lo + hi) >> 1;
      if (buf[src][mid] > a) { res = mid; hi = mid - 1; }
      else lo = mid + 1;
    }
    tok[a] = res;
  }
}

// ---------------------------------------------------------------------------
// Kernel C: cl_out = cl + s[tok];  ql = cl_out + rl @ W_r + b_r
// 4096*128 elements, fully coalesced.
// ---------------------------------------------------------------------------
__global__ __launch_bounds__(256) void atom_kernel(
    const float* __restrict__ cl, const float* __restrict__ rl,
    const float* __restrict__ W_r, const float* __restrict__ b_r,
    const float* __restrict__ s, const int* __restrict__ tok,
    float* __restrict__ cl_out, float* __restrict__ ql_out) {
  const int idx = blockIdx.x * 256 + threadIdx.x;
  const int a = idx >> 7;
  const int c = idx & 127;
  const int t = tok[a];
  const float sv  = s[(size_t)t * C_ATOM + c];
  const float co  = cl[idx] + sv;
  const float r0 = rl[a * 3 + 0], r1 = rl[a * 3 + 1], r2 = rl[a * 3 + 2];
  const float q = co + r0 * W_r[c] + r1 * W_r[128 + c] + r2 * W_r[256 + c] + b_r[c];
  cl_out[idx] = co;
  ql_out[idx] = q;
}

// ---------------------------------------------------------------------------
// Kernel D: plm_out = plm + gather(LN(zij)@W_z + b_z) * valid
// One workgroup per block b (128 WGs, 256 threads = 8 waves).
//   unique token pairs per block: i_local = q>>2 (8), j_local = k>>2 (32)
//   => [256 x 128] @ [128 x 16] fp32 GEMM via v_wmma_f32_16x16x4_f32.
// ---------------------------------------------------------------------------
#define D_AST 131  // 128 + 3 (odd stride for conflict-free A reads)
#define D_ZST 17   // zres row stride

__global__ __launch_bounds__(256) void pair_kernel(
    const float* __restrict__ zij, const float* __restrict__ ln_z_w,
    const float* __restrict__ W_z, const float* __restrict__ b_z,
    const float* __restrict__ plm, float* __restrict__ plm_out) {
  extern __shared__ float smem[];
  float* Alds = smem;                       // 256 * 131 floats
  float* Wz   = smem + 256 * D_AST;         // 128 * 16 floats
  float* zres = Wz + C_Z * C_PAIR;          // 256 * 17 floats

  const int tid  = threadIdx.x;
  const int wave = tid >> 5;
  const int lane = tid & 31;
  const int half = lane >> 4;
  const int l16  = lane & 15;
  const int blk  = blockIdx.x;

  // Stage W_z in LDS
  for (int i = tid; i < C_Z * C_PAIR; i += 256) Wz[i] = W_z[i];

  // Phase 1: LN the 256 needed (i,j) token-pair rows into LDS (wave-per-row)
  const float4 wv = *(const float4*)(ln_z_w + lane * 4);
  for (int it = 0; it < 32; ++it) {
    const int r  = wave * 32 + it;
    const int il = r >> 5;            // i_local 0..7
    const int jl = r & 31;            // j_local 0..31
    const int itok = blk * 8 + il;
    const int jtok = blk * 8 - 12 + jl;
    float* dst = Alds + r * D_AST + lane * 4;
    if (jtok >= 0 && jtok < N_TOK) {
      const float4 x = *(const float4*)(zij +
          ((size_t)itok * N_TOK + (size_t)jtok) * C_Z + lane * 4);
      float s  = x.x + x.y + x.z + x.w;
      float ss = x.x*x.x + x.y*x.y + x.z*x.z + x.w*x.w;
      s  = wave_sum(s);
      ss = wave_sum(ss);
      const float mu   = s * (1.f / C_Z);
      const float var  = ss * (1.f / C_Z) - mu * mu;
      const float rstd = rsqrtf(var + EPSV);
      dst[0] = (x.x - mu) * rstd * wv.x;
      dst[1] = (x.y - mu) * rstd * wv.y;
      dst[2] = (x.z - mu) * rstd * wv.z;
      dst[3] = (x.w - mu) * rstd * wv.w;
    } else {
      dst[0] = 0.f; dst[1] = 0.f; dst[2] = 0.f; dst[3] = 0.f;
    }
  }
  __syncthreads();

  // Phase 2: per-wave [32 x 128] @ [128 x 16] -> two 16x16 tiles
  const int Mb = wave * 32;
  v8f acc0 = {0.f, 0.f, 0.f, 0.f, 0.f, 0.f, 0.f, 0.f};
  v8f acc1 = {0.f, 0.f, 0.f, 0.f, 0.f, 0.f, 0.f, 0.f};
  for (int k0 = 0; k0 < C_Z; k0 += 4) {
    const int kk = k0 + 2 * half;
    v2f bb;
    bb.x = Wz[kk * C_PAIR + l16];
    bb.y = Wz[(kk + 1) * C_PAIR + l16];
    v2f a0, a1;
    a0.x = Alds[(Mb + l16) * D_AST + kk];
    a0.y = Alds[(Mb + l16) * D_AST + kk + 1];
    a1.x = Alds[(Mb + 16 + l16) * D_AST + kk];
    a1.y = Alds[(Mb + 16 + l16) * D_AST + kk + 1];
    acc0 = __builtin_amdgcn_wmma_f32_16x16x4_f32(false, a0, false, bb,
                                                 (short)0, acc0, false, false);
    acc1 = __builtin_amdgcn_wmma_f32_16x16x4_f32(false, a1, false, bb,
                                                 (short)0, acc1, false, false);
  }
  const float bz = b_z[l16];
#pragma unroll
  for (int v = 0; v < 8; ++v) {
    const int m = v + half * 8;
    zres[(Mb + m) * D_ZST + l16]      = acc0[v] + bz;
    zres[(Mb + 16 + m) * D_ZST + l16] = acc1[v] + bz;
  }
  __syncthreads();

  // Phase 3: broadcast 4x4 into plm tile with valid mask (coalesced)
  const size_t base = (size_t)blk * (N_QUERY * N_KEY * C_PAIR);
  for (int i = tid; i < N_QUERY * N_KEY * C_PAIR; i += 256) {
    const int q   = i >> 11;          // /(128*16)
    const int rem = i & 2047;
    const int k   = rem >> 4;
    const int c   = rem & 15;
    const int kidx = blk * 32 - 48 + k;
    float v = 0.f;
    if (kidx >= 0 && kidx < N_ATOM) {
      const int r = (q >> 2) * 32 + (k >> 2);
      v = zres[r * D_ZST + c];
    }
    plm_out[base + i] = plm[base + i] + v;
  }
}

// ---------------------------------------------------------------------------
// Launch
// ---------------------------------------------------------------------------
extern "C" void kernel_launch(void* const* d_in, const int* in_sizes, int n_in,
                              void* d_out, int out_size, void* d_ws, size_t ws_size,
                              hipStream_t stream) {
  const float* token_mask = (const float*)d_in[0];
  const int*   napt       = (const int*)d_in[1];
  const float* cl         = (const float*)d_in[2];
  const float* plm        = (const float*)d_in[3];
  const float* si_trunk   = (const float*)d_in[4];
  const float* zij_trunk  = (const float*)d_in[5];
  const float* rl         = (const float*)d_in[6];
  const float* ln_s_w     = (const float*)d_in[7];
  const float* W_s        = (const float*)d_in[8];
  const float* b_s        = (const float*)d_in[9];
  const float* ln_z_w     = (const float*)d_in[10];
  const float* W_z        = (const float*)d_in[11];
  const float* b_z        = (const float*)d_in[12];
  const float* W_r        = (const float*)d_in[13];
  const float* b_r        = (const float*)d_in[14];

  // Workspace: s (1024x128 f32), tok (4096 i32)
  float* sws  = (float*)d_ws;
  int*   tokp = (int*)((char*)d_ws + (size_t)N_TOK * C_ATOM * sizeof(float));

  // Outputs concatenated: cl_out (524288), plm_out (8388608), ql (524288)
  float* cl_out  = (float*)d_out;
  float* plm_out = cl_out + (size_t)N_ATOM * C_ATOM;
  float* ql_out  = plm_out + (size_t)N_BLOCKS * N_QUERY * N_KEY * C_PAIR;

  s_proj_kernel<<<N_TOK / 32, 256, 0, stream>>>(si_trunk, ln_s_w, W_s, b_s,
                                                token_mask, sws);
  tok_kernel<<<1, 1024, 0, stream>>>(napt, tokp);
  atom_kernel<<<(N_ATOM * C_ATOM) / 256, 256, 0, stream>>>(
      cl, rl, W_r, b_r, sws, tokp, cl_out, ql_out);

  const size_t shmem =
      (size_t)(256 * D_AST + C_Z * C_PAIR + 256 * D_ZST) * sizeof(float);
  pair_kernel<<<N_BLOCKS, 256, shmem, stream>>>(zij_trunk, ln_z_w, W_z, b_z,
                                                plm, plm_out);
}